// CodeBook_70961449664874
// MI455X (gfx1250) — compile-verified
//
#include <hip/hip_runtime.h>
#include <hip/hip_bf16.h>

typedef float v2f __attribute__((ext_vector_type(2)));
typedef float v4f __attribute__((ext_vector_type(4)));
typedef float v8f __attribute__((ext_vector_type(8)));

#define LATENT 64
#define NCODES 1024
#define ZQ_ELEMS (16u * 64u * 64u * 64u)   // 4,194,304 = B*C*H*W
#define N_IDX    65536u                    // B*H*W
#define TOTAL4   (ZQ_ELEMS / 4u)           // 1,048,576 float4s
#define BLOCKS_B 512
#define THREADS_B 256
#define STRIDE_B ((unsigned)(BLOCKS_B * THREADS_B))
#define ITERS_B  (TOTAL4 / STRIDE_B)       // exactly 8 — uniform trip count

// ---------------------------------------------------------------------------
// Kernel 1: k* = argmin_k ||codebook[k]||^2  (first-min tie-break like argmin)
// Encoded as u64 key (float bits << 32) | k; s >= 0 keeps float-bit ordering.
// ---------------------------------------------------------------------------
__global__ void vq_kstar_kernel(const float* __restrict__ cb,
                                int* __restrict__ ws_kstar) {
    __shared__ unsigned long long best;
    const int k = threadIdx.x;            // 1024 threads, one per code
    if (k == 0) best = ~0ULL;
    __syncthreads();

    const v4f* row = (const v4f*)(cb + k * LATENT);
    float s = 0.0f;
#pragma unroll
    for (int j = 0; j < LATENT / 4; ++j) {
        v4f q = row[j];
        s = fmaf(q.x, q.x, s);
        s = fmaf(q.y, q.y, s);
        s = fmaf(q.z, q.z, s);
        s = fmaf(q.w, q.w, s);
    }
    unsigned long long key =
        ((unsigned long long)__float_as_uint(s) << 32) | (unsigned)k;
    atomicMin(&best, key);                // ds_min_u64
    __syncthreads();
    if (k == 0) ws_kstar[0] = (int)(best & 0xFFFFFFFFULL);
}

// ---------------------------------------------------------------------------
// Kernel 2: stream z (NT loads) -> broadcast z_q (NT stores); accumulate
// sum((v[c]-z)^2) on the matrix pipe via V_WMMA_F32_16X16X4_F32 with an
// all-ones B matrix. Trip count is compile-time (8), fully unrolled: no
// divergence, EXEC provably all-ones at every WMMA. Deterministic partials.
// ---------------------------------------------------------------------------
__global__ void vq_main_kernel(const float* __restrict__ z,
                               float* __restrict__ zq,
                               const int* __restrict__ ws_kstar,
                               float* __restrict__ partials) {
    __shared__ float vsh[LATENT];
    __shared__ float wavesum[THREADS_B / 32];

    const int kstar = ws_kstar[0];
    if (threadIdx.x < LATENT) {
        // v[c] = z[0, c, h*, w*] = z[c*4096 + kstar]
        vsh[threadIdx.x] = z[threadIdx.x * 4096 + kstar];
    }
    __syncthreads();

    const v4f* __restrict__ z4 = (const v4f*)z;
    v4f* __restrict__ o4 = (v4f*)zq;

    v8f acc = {};                             // 16x16 f32 accumulator, zero
    v2f ones; ones.x = 1.0f; ones.y = 1.0f;   // B = 4x16 all-ones

    const unsigned tid = blockIdx.x * THREADS_B + threadIdx.x;
#pragma unroll
    for (unsigned it = 0; it < ITERS_B; ++it) {
        const unsigned i4 = tid + it * STRIDE_B;
        if (it + 1 < ITERS_B)                 // compile-time after unroll
            __builtin_prefetch(&z4[i4 + STRIDE_B], 0, 1); // global_prefetch_b8

        v4f q = __builtin_nontemporal_load(&z4[i4]);      // b128, TH=NT
        const float vc = vsh[(i4 >> 10) & (LATENT - 1)];  // c = (4*i4/4096)%64
        const float d0 = q.x - vc, d1 = q.y - vc;
        const float d2 = q.z - vc, d3 = q.w - vc;

        v2f a;
        a.x = fmaf(d0, d0, d1 * d1);
        a.y = fmaf(d2, d2, d3 * d3);
        // D[m,n] += sum_k A[m,k] * 1 : every A slot counted once per column.
        acc = __builtin_amdgcn_wmma_f32_16x16x4_f32(
            false, a, false, ones, (short)0, acc, false, false);

        v4f w; w.x = vc; w.y = vc; w.z = vc; w.w = vc;
        __builtin_nontemporal_store(w, &o4[i4]);          // b128, TH=NT
    }

    // Lanes 0..15 hold rows 0..7 in acc[0..7]; lanes 16..31 hold rows 8..15.
    float t = acc[0] + acc[1] + acc[2] + acc[3] +
              acc[4] + acc[5] + acc[6] + acc[7];
    t += __shfl_xor(t, 16, 32);          // every lane now has the wave total

    const int lane = threadIdx.x & 31;
    const int wave = threadIdx.x >> 5;
    if (lane == 0) wavesum[wave] = t;
    __syncthreads();
    if (threadIdx.x == 0) {
        float s = 0.0f;
#pragma unroll
        for (int i = 0; i < THREADS_B / 32; ++i) s += wavesum[i];
        partials[blockIdx.x] = s;
    }
}

// ---------------------------------------------------------------------------
// Kernel 3: block 0 reduces the 512 block partials -> loss; blocks 1..64
// broadcast-store the constant index to the 65536-entry index output.
// ---------------------------------------------------------------------------
__global__ void vq_tail_kernel(float* __restrict__ out,
                               const float* __restrict__ partials,
                               const int* __restrict__ ws_kstar) {
    if (blockIdx.x == 0) {
        __shared__ float sh[256];
        float s = partials[threadIdx.x] + partials[threadIdx.x + 256];
        sh[threadIdx.x] = s;
        __syncthreads();
        for (int off = 128; off > 0; off >>= 1) {
            if (threadIdx.x < off) sh[threadIdx.x] += sh[threadIdx.x + off];
            __syncthreads();
        }
        if (threadIdx.x == 0) {
            // loss = (1 + beta) * mean((z_q - z)^2), beta = 0.25
            out[ZQ_ELEMS + N_IDX] = 1.25f * sh[0] / (float)ZQ_ELEMS;
        }
    } else {
        const float kf = (float)ws_kstar[0];
        const unsigned i4 = (blockIdx.x - 1) * blockDim.x + threadIdx.x;
        v4f* p = (v4f*)(out + ZQ_ELEMS);   // indices region, 16B aligned
        v4f w; w.x = kf; w.y = kf; w.z = kf; w.w = kf;
        p[i4] = w;                         // 64 blocks * 256 * 4 = 65536
    }
}

extern "C" void kernel_launch(void* const* d_in, const int* in_sizes, int n_in,
                              void* d_out, int out_size, void* d_ws, size_t ws_size,
                              hipStream_t stream) {
    const float* z  = (const float*)d_in[0];   // [16, 64, 64, 64] f32
    const float* cb = (const float*)d_in[1];   // [1024, 64] f32
    float* out = (float*)d_out;                // z_q | indices | loss

    int*   ws_kstar = (int*)d_ws;
    float* partials = (float*)((char*)d_ws + 256);   // 512 floats

    vq_kstar_kernel<<<1, NCODES, 0, stream>>>(cb, ws_kstar);
    vq_main_kernel<<<BLOCKS_B, THREADS_B, 0, stream>>>(z, out, ws_kstar, partials);
    vq_tail_kernel<<<1 + (N_IDX / 4 / THREADS_B), THREADS_B, 0, stream>>>(
        out, partials, ws_kstar);
}